// CafeEmbeddingBagCollection_6597069767063
// MI455X (gfx1250) — compile-verified
//
#include <hip/hip_runtime.h>
#include <stdint.h>

// ---------------------------------------------------------------------------
// CafeEmbeddingBagCollection pooled-embedding gather on MI455X (gfx1250).
//
// Memory-bound: ~108 MB of traffic -> ~4.6 us floor at 23.3 TB/s. No matmul
// (zero matrix FLOPs), so no WMMA; the CDNA5 feature that matters here is the
// async global->LDS DMA path (GLOBAL_LOAD_ASYNC_TO_LDS_B128 / ASYNCcnt):
// one instruction gathers a full 512 B embedding row (32 lanes x 16 B), and a
// triple-buffered per-wave pipeline keeps 15 rows (7.5 KB) in flight per wave
// -> ~30 MB aggregate across 4096 waves, above the ~18 MB bandwidth-delay
// product of 23.3 TB/s x ~800 ns.
// ---------------------------------------------------------------------------

#define DIM 128              // embedding dim (floats)
#define ROW_FLOATS DIM       // 512 bytes per row
#define CHUNK 5              // rows per pipeline stage (L=50 -> 10 stages)
#define NBUF 3               // triple buffering
#define WAVES_PER_BLOCK 8    // 256-thread blocks, wave32

typedef int v4i __attribute__((ext_vector_type(4)));
typedef __attribute__((address_space(1))) v4i global_v4i;
typedef __attribute__((address_space(3))) v4i lds_v4i;

__device__ __forceinline__ void async_row_ld_b128(void* lds_dst, const void* gsrc) {
  // Bit-exact pointer re-interpretation:
  //  - generic global address == addrspace(1) address
  //  - generic LDS address low 32 bits == LDS byte offset == addrspace(3) value
  global_v4i* gp = (global_v4i*)(uintptr_t)gsrc;
  lds_v4i*    lp = (lds_v4i*)(uint32_t)(uintptr_t)lds_dst;
#if defined(__has_builtin) && __has_builtin(__builtin_amdgcn_global_load_async_to_lds_b128)
  __builtin_amdgcn_global_load_async_to_lds_b128(gp, lp, /*offset=*/0, /*cpol=*/0);
#else
  asm volatile("global_load_async_to_lds_b128 %0, %1, off"
               :: "v"((uint32_t)(uintptr_t)lp), "v"(gp)
               : "memory");
#endif
}

template <int N>
__device__ __forceinline__ void wait_asynccnt_le() {
#if defined(__has_builtin) && __has_builtin(__builtin_amdgcn_s_wait_asynccnt)
  __builtin_amdgcn_s_wait_asynccnt(N);
#else
  asm volatile("s_wait_asynccnt %0" :: "i"(N) : "memory");
#endif
  // Compiler fence: LDS reads below must not be hoisted above the wait.
  asm volatile("" ::: "memory");
}

// Issue CHUNK async row-gathers (one 512 B row per instruction; lane l moves
// bytes [16l, 16l+16)). Index math is wave-uniform -> readfirstlane pins it to
// SGPRs/SALU. Runtime '%' replaced by host-precomputed magic reciprocal.
__device__ __forceinline__ void issue_chunk(const int* __restrict__ qres,
                                            const float* __restrict__ hotW,
                                            const float* __restrict__ hashW,
                                            int start, int len, int cbase,
                                            int hotRows, uint32_t hashRows,
                                            uint32_t magic, int shift,
                                            float* lds_chunk, int lane) {
#pragma unroll
  for (int r = 0; r < CHUNK; ++r) {
    int t = cbase + r;
    if (t > len - 1) t = len - 1;        // clamp tail: duplicate loads, masked in accumulate
    int q = __builtin_amdgcn_readfirstlane(qres[start + t]);   // wave-uniform -> SGPR
    const float* row;
    if (q < 0) {
      int hi = -q;
      if (hi > hotRows - 1) hi = hotRows - 1;
      row = hotW + (size_t)hi * DIM;
    } else {
      uint32_t qu = (uint32_t)q;
      uint32_t rem;
      if (magic != 0) {
        // est = floor(qu / hashRows) or one less; single correction suffices
        // for qu < 2^31 (magic = floor(2^(32+shift)/d), shift = floor(log2 d)).
        uint32_t est = (uint32_t)(((uint64_t)qu * magic) >> 32) >> shift;
        rem = qu - est * hashRows;
        if (rem >= hashRows) rem -= hashRows;
      } else {
        rem = qu & (hashRows - 1u);      // power-of-two table size
      }
      row = hashW + (size_t)rem * DIM;
    }
    async_row_ld_b128(lds_chunk + r * ROW_FLOATS + lane * 4, row + lane * 4);
  }
}

__global__ __launch_bounds__(WAVES_PER_BLOCK * 32)
void cafe_embbag_pool_kernel(const int* __restrict__ offsets,
                             const int* __restrict__ qres,
                             const float* __restrict__ hotW,
                             const float* __restrict__ hashW,
                             float* __restrict__ out,
                             int B, int N, int hotRows, uint32_t hashRows,
                             uint32_t magic, int shift) {
  // Per-wave triple buffer: 3 * CHUNK * 512 B = 7.5 KB; 8 waves -> 60 KB/block
  // (5 blocks per 320 KB WGP).
  __shared__ float smem[WAVES_PER_BLOCK * NBUF * CHUNK * ROW_FLOATS];

  const int lane = threadIdx.x & 31;
  const int wave = threadIdx.x >> 5;
  const int bag  = blockIdx.x * WAVES_PER_BLOCK + wave;   // wave-uniform
  if (bag >= B) return;

  const int start = offsets[bag];
  const int end   = (bag + 1 < B) ? offsets[bag + 1] : N;
  const int len   = end - start;

  float4 acc = make_float4(0.f, 0.f, 0.f, 0.f);
  float* buf = &smem[wave * NBUF * CHUNK * ROW_FLOATS];
  const int CS = CHUNK * ROW_FLOATS;

  if (len > 0) {
    const int nch = (len + CHUNK - 1) / CHUNK;

    // Prologue: fill stages 0 and 1.
    issue_chunk(qres, hotW, hashW, start, len, 0, hotRows, hashRows,
                magic, shift, buf + 0 * CS, lane);
    if (nch > 1)
      issue_chunk(qres, hotW, hashW, start, len, CHUNK, hotRows, hashRows,
                  magic, shift, buf + 1 * CS, lane);

    for (int c = 0; c < nch; ++c) {
      if (c + 2 < nch)
        issue_chunk(qres, hotW, hashW, start, len, (c + 2) * CHUNK,
                    hotRows, hashRows, magic, shift,
                    buf + ((c + 2) % NBUF) * CS, lane);

      // In-order async completion: if any load of chunk c is outstanding, all
      // loads of chunks c+1/c+2 are too, so cnt > threshold. Therefore:
      if (c + 2 < nch)       wait_asynccnt_le<2 * CHUNK>();  // c+1, c+2 in flight
      else if (c + 1 < nch)  wait_asynccnt_le<CHUNK>();      // only c+1 in flight
      else                   wait_asynccnt_le<0>();          // drain

      // Accumulate chunk c from LDS (ds_load_b128 per row per lane).
      float* cur = buf + (c % NBUF) * CS;
      const int cbase = c * CHUNK;
#pragma unroll
      for (int r = 0; r < CHUNK; ++r) {
        if (cbase + r < len) {
          float4 v = *(const float4*)(cur + r * ROW_FLOATS + lane * 4);
          acc.x += v.x; acc.y += v.y; acc.z += v.z; acc.w += v.w;
        }
      }
    }
  }

  // One global_store_b128 per lane: out[bag, lane*4 .. lane*4+3].
  *(float4*)(out + (size_t)bag * DIM + lane * 4) = acc;
}

extern "C" void kernel_launch(void* const* d_in, const int* in_sizes, int n_in,
                              void* d_out, int out_size, void* d_ws, size_t ws_size,
                              hipStream_t stream) {
  // setup_inputs() order:
  //   0: feature_ids [N] int32   (unused; routing fully encoded in query_results)
  //   1: offsets     [B] int32
  //   2: query_results [N] int32
  //   3: hot_W  [(HOT+1)*128] f32
  //   4: hash_W [HASH*128]    f32
  const int*   offsets = (const int*)d_in[1];
  const int*   qres    = (const int*)d_in[2];
  const float* hotW    = (const float*)d_in[3];
  const float* hashW   = (const float*)d_in[4];
  float*       out     = (float*)d_out;

  const int N        = in_sizes[0];
  const int B        = in_sizes[1];
  const int hotRows  = in_sizes[3] / DIM;
  const uint32_t d   = (uint32_t)(in_sizes[4] / DIM);   // hashRows

  // Host-side magic reciprocal for q % d (q < 2^31). For power-of-two d use
  // the mask path (magic == 0 sentinel).
  uint32_t magic = 0;
  int shift = 0;
  if (d > 1u && (d & (d - 1u)) != 0u) {
    shift = 31 - __builtin_clz(d);                       // floor(log2 d)
    magic = (uint32_t)((1ull << (32 + shift)) / d);      // < 2^32 since d not pow2
  }

  const int blocks = (B + WAVES_PER_BLOCK - 1) / WAVES_PER_BLOCK;
  hipLaunchKernelGGL(cafe_embbag_pool_kernel, dim3(blocks),
                     dim3(WAVES_PER_BLOCK * 32), 0, stream,
                     offsets, qres, hotW, hashW, out, B, N, hotRows, d,
                     magic, shift);
}